// SGP_LAYER_89713276878973
// MI455X (gfx1250) — compile-verified
//
#include <hip/hip_runtime.h>
#include <hip/hip_bf16.h>

// ---- problem constants (from reference setup) ----
#define NB   2      // batch B
#define NL   192    // sequence L
#define NH   8      // heads H
#define NHD  512    // hdim
#define NV   64     // vdim
#define NM   64     // inducing M
#define NS   4      // samples S
#define GP_JITTER 0.001f
#define BIGMOD (1<<30)
#define KL_OFS (NB*NS*NL*NHD)   // 786432

typedef float v2f __attribute__((ext_vector_type(2)));
typedef float v8f __attribute__((ext_vector_type(8)));

// Fragment loader: T=1 -> operand stored transposed (stride ld between the two
// k-elements); T=0 -> contiguous pair, single 8-byte vector load.
template<int T>
__device__ __forceinline__ v2f load_frag(const float* __restrict__ p, long ld, long k)
{
    v2f r;
    if (T) { r[0] = p[k * ld]; r[1] = p[k * ld + ld]; }
    else   { r = *(const v2f*)(p + k); }
    return r;
}

// =====================================================================
// Generic batched WMMA fp32 GEMM:  C = alpha * opA(A) * opB(B) + beta * Cadd
// opA(A) logical (M x K), opB(B) logical (K x N).  One 16x16 C tile / wave.
// Batch offsets: base + ((batch/div) % mod) * stride   (per operand).
// K unrolled x16: 8 overlapped fragment loads feeding 4 chained
// V_WMMA_F32_16X16X4_F32 issues per trip.
// =====================================================================
template<int TA, int TB>
__global__ void wmma_gemm(const float* __restrict__ A, int lda, long sA, int adiv, int amod,
                          const float* __restrict__ B, int ldb, long sB, int bdiv, int bmod,
                          float* __restrict__ C, int ldc, long sC,
                          const float* __restrict__ Cadd, long sCadd, int cadiv, int camod, float beta,
                          int M, int N, int K, float alpha)
{
    int bat   = blockIdx.y;
    int lane  = threadIdx.x & 31;
    int wv    = threadIdx.x >> 5;
    int tilesX = N >> 4;
    int tilesY = M >> 4;
    int tile  = blockIdx.x * (blockDim.x >> 5) + wv;
    if (tile >= tilesX * tilesY) return;           // wave-uniform: EXEC stays all-ones for WMMA
    int m0 = (tile / tilesX) << 4;
    int n0 = (tile % tilesX) << 4;

    const float* Ab = A + (long)((bat / adiv) % amod) * sA;
    const float* Bb = B + (long)((bat / bdiv) % bmod) * sB;
    float*       Cb = C + (long)bat * sC;

    int r  = lane & 15;           // row (A) / col (B) within tile
    int kp = (lane >> 4) << 1;    // k-pair base per 16x4 fp32 A/B layout

    // per-lane fragment base pointers (kp folded in)
    const float* ap = TA ? (Ab + (long)kp * lda + (m0 + r)) : (Ab + (long)(m0 + r) * lda + kp);
    const float* bp = TB ? (Bb + (long)(n0 + r) * ldb + kp) : (Bb + (long)kp * ldb + (n0 + r));
    // for the contiguous path, k advances the pointer by k elements;
    // for the strided path, by k*ld elements -> handled inside load_frag.
    const long aK = TA ? 1 : 1;  (void)aK;

    v8f acc = {0.f,0.f,0.f,0.f,0.f,0.f,0.f,0.f};
    for (int k0 = 0; k0 < K; k0 += 16) {
        v2f a0 = load_frag<TA>(ap, lda, k0 +  0);
        v2f a1 = load_frag<TA>(ap, lda, k0 +  4);
        v2f a2 = load_frag<TA>(ap, lda, k0 +  8);
        v2f a3 = load_frag<TA>(ap, lda, k0 + 12);
        v2f b0 = load_frag<TB>(bp, ldb, k0 +  0);
        v2f b1 = load_frag<TB>(bp, ldb, k0 +  4);
        v2f b2 = load_frag<TB>(bp, ldb, k0 +  8);
        v2f b3 = load_frag<TB>(bp, ldb, k0 + 12);
        acc = __builtin_amdgcn_wmma_f32_16x16x4_f32(false, a0, false, b0, (short)0, acc, false, false);
        acc = __builtin_amdgcn_wmma_f32_16x16x4_f32(false, a1, false, b1, (short)0, acc, false, false);
        acc = __builtin_amdgcn_wmma_f32_16x16x4_f32(false, a2, false, b2, (short)0, acc, false, false);
        acc = __builtin_amdgcn_wmma_f32_16x16x4_f32(false, a3, false, b3, (short)0, acc, false, false);
    }

    const float* Db = Cadd ? (Cadd + (long)((bat / cadiv) % camod) * sCadd) : nullptr;
    int mofs = (lane >> 4) << 3;      // lanes 16-31 hold rows M+8 per C/D layout
#pragma unroll
    for (int rr = 0; rr < 8; rr++) {
        int m = m0 + rr + mofs;
        int n = n0 + (lane & 15);
        float v = alpha * acc[rr];
        if (Db) v += beta * Db[(long)m * ldc + n];
        Cb[(long)m * ldc + n] = v;
    }
}

// =====================================================================
// Feature gather + ARD scaling.  qs/vg laid out (h,b,l,d); qn = row norms.
// =====================================================================
__global__ void k_feat_q(const float* __restrict__ qall, const float* __restrict__ vgall,
                         const float* __restrict__ mask, const float* __restrict__ log_ls,
                         float* __restrict__ qs, float* __restrict__ vg, float* __restrict__ qn)
{
    int i = blockIdx.x * blockDim.x + threadIdx.x;
    if (i >= NH * NB * NL) return;
    int h = i / (NB * NL);
    int rem = i - h * NB * NL;
    int b = rem / NL, l = rem - b * NL;
    const float* src  = qall  + ((long)(b * NL + l)) * NHD + h * NV;
    const float* vsrc = vgall + ((long)(b * NL + l)) * NHD + h * NV;
    float m = mask[b * NL + l];
    float* qd = qs + (long)i * NV;
    float* vd = vg + (long)i * NV;
    float s = 0.f;
    for (int d = 0; d < NV; d++) {
        float sc = expf(-log_ls[h * NV + d]);
        float v = src[d] * sc;
        qd[d] = v; s += v * v;
        vd[d] = vsrc[d] * m;
    }
    qn[i] = s;
}

__global__ void k_feat_k(const float* __restrict__ kball, const float* __restrict__ log_ls,
                         float* __restrict__ ks, float* __restrict__ kn)
{
    int i = blockIdx.x * blockDim.x + threadIdx.x;
    if (i >= NH * NM) return;
    int h = i / NM;
    const float* src = kball + (long)i * NV;
    float* kd = ks + (long)i * NV;
    float s = 0.f;
    for (int d = 0; d < NV; d++) {
        float v = src[d] * expf(-log_ls[h * NV + d]);
        kd[d] = v; s += v * v;
    }
    kn[i] = s;
}

// =====================================================================
// Turn Gram matrices into ARD kernels (in-place): K = sf*exp(-0.5*(n1+n2-2G))*mask
// =====================================================================
__global__ void k_make_qq(float* __restrict__ G, const float* __restrict__ qn,
                          const float* __restrict__ mask, const float* __restrict__ log_sf)
{
    long i = (long)blockIdx.x * blockDim.x + threadIdx.x;
    if (i >= (long)NH * NB * NL * NL) return;
    int hb = (int)(i / (NL * NL));
    int rem = (int)(i - (long)hb * NL * NL);
    int l = rem / NL, k = rem - l * NL;
    int h = hb >> 1, b = hb & 1;
    float sf = expf(log_sf[h]);
    float d2 = qn[hb * NL + l] + qn[hb * NL + k] - 2.f * G[i];
    G[i] = sf * expf(-0.5f * d2) * mask[b * NL + l] * mask[b * NL + k];
}

__global__ void k_make_bg(float* __restrict__ G, const float* __restrict__ kn,
                          const float* __restrict__ qn, const float* __restrict__ mask,
                          const float* __restrict__ log_sf)
{
    long i = (long)blockIdx.x * blockDim.x + threadIdx.x;
    if (i >= (long)NH * NB * NM * NL) return;
    int hb = (int)(i / (NM * NL));
    int rem = (int)(i - (long)hb * NM * NL);
    int m = rem / NL, l = rem - m * NL;
    int h = hb >> 1, b = hb & 1;
    float sf = expf(log_sf[h]);
    float d2 = kn[h * NM + m] + qn[hb * NL + l] - 2.f * G[i];
    G[i] = sf * expf(-0.5f * d2) * mask[b * NL + l];
}

__global__ void k_make_bb(float* __restrict__ G, const float* __restrict__ kn,
                          const float* __restrict__ log_sf)
{
    int i = blockIdx.x * blockDim.x + threadIdx.x;
    if (i >= NH * NM * NM) return;
    int h = i / (NM * NM);
    int rem = i - h * NM * NM;
    int m = rem / NM, n = rem - m * NM;
    float sf = expf(log_sf[h]);
    float d2 = kn[h * NM + m] + kn[h * NM + n] - 2.f * G[i];
    G[i] = sf * expf(-0.5f * d2);
}

// =====================================================================
// 64x64 Cholesky (one block / head, thread = row)
// =====================================================================
__global__ void k_chol64(const float* __restrict__ Kbb, float* __restrict__ chol)
{
    __shared__ float a[NM][NM + 1];
    int h = blockIdx.x, t = threadIdx.x;
    for (int j = 0; j < NM; j++)
        a[t][j] = Kbb[(long)h * NM * NM + t * NM + j] + (t == j ? GP_JITTER : 0.f);
    __syncthreads();
    for (int j = 0; j < NM; j++) {
        if (t == j) a[j][j] = sqrtf(a[j][j]);
        __syncthreads();
        if (t > j) a[t][j] /= a[j][j];
        __syncthreads();
        if (t > j) {
            float f = a[t][j];
            for (int k = j + 1; k <= t; k++) a[t][k] -= f * a[k][j];
        }
        __syncthreads();
    }
    for (int j = 0; j < NM; j++)
        chol[(long)h * NM * NM + t * NM + j] = (j <= t) ? a[t][j] : 0.f;
}

// =====================================================================
// Forward triangular solve: chol(64x64, per head) * X = RHS(64 x NL), per (h,b).
// grid.x = h*B+b, grid.y = column chunk of 64, thread = column.
// =====================================================================
__global__ void k_trisolve(const float* __restrict__ chol, const float* __restrict__ rhs,
                           float* __restrict__ v1)
{
    __shared__ float a[NM][NM + 1];
    __shared__ float x[NM][NM + 1];
    int hb = blockIdx.x, h = hb >> 1;
    int t = threadIdx.x;
    int col = blockIdx.y * NM + t;
    for (int j = 0; j < NM; j++) a[j][t] = chol[(long)h * NM * NM + j * NM + t];
    __syncthreads();
    const float* R = rhs + (long)hb * NM * NL;
    for (int m = 0; m < NM; m++) {
        float s = R[m * NL + col];
        for (int k = 0; k < m; k++) s -= a[m][k] * x[k][t];
        x[m][t] = s / a[m][m];
    }
    float* V = v1 + (long)hb * NM * NL;
    for (int m = 0; m < NM; m++) V[m * NL + col] = x[m][t];
}

__global__ void k_maskv2(const float* __restrict__ v1, const float* __restrict__ mask,
                         float* __restrict__ v2)
{
    long i = (long)blockIdx.x * blockDim.x + threadIdx.x;
    if (i >= (long)NH * NB * NM * NL) return;
    int hb = (int)(i / (NM * NL));
    int l = (int)(i % NL);
    int b = hb & 1;
    v2[i] = v1[i] * mask[b * NL + l];
}

// =====================================================================
// s_local = diag(exp(lssd)) + tril(ltri, -1)   (h,d,m,n)
// =====================================================================
__global__ void k_sloc(const float* __restrict__ ltri, const float* __restrict__ lssd,
                       float* __restrict__ sloc)
{
    long i = (long)blockIdx.x * blockDim.x + threadIdx.x;
    if (i >= (long)NH * NV * NM * NM) return;
    int n = (int)(i % NM);
    int m = (int)((i / NM) % NM);
    long hd = i / (NM * NM);
    float v;
    if (m == n)      v = expf(lssd[hd * NM + m]);
    else if (m > n)  v = ltri[i];
    else             v = 0.f;
    sloc[i] = v;
}

// diag += jitter + (1 - mask^2)  (mask_cov term), per (b,d) matrix of one head
__global__ void k_prepdiag(float* __restrict__ cov, const float* __restrict__ mask)
{
    int mat = blockIdx.x;          // b*NV + d
    int l   = threadIdx.x;         // 0..NL-1
    int b   = mat >> 6;
    float m = mask[b * NL + l];
    float* A = cov + (long)mat * NL * NL;
    A[l * NL + l] += GP_JITTER + (1.f - m * m);
}

// =====================================================================
// In-place 192x192 Cholesky, one block per matrix, LDS-cached pivot column.
// =====================================================================
__global__ void k_chol192(float* __restrict__ cov)
{
    __shared__ float colj[NL];
    float* A = cov + (long)blockIdx.x * NL * NL;
    int t = threadIdx.x;   // 0..NL-1
    for (int j = 0; j < NL; j++) {
        if (t == 0) A[j * NL + j] = sqrtf(A[j * NL + j]);
        __syncthreads();
        float d = A[j * NL + j];
        if (t > j) A[t * NL + j] /= d;
        __syncthreads();
        colj[t] = (t > j) ? A[t * NL + j] : 0.f;
        __syncthreads();
        if (t > j) {
            float f = colj[t];
            for (int k = j + 1; k <= t; k++) A[t * NL + k] -= f * colj[k];
        }
        __syncthreads();
    }
}

// zero upper triangle + multiply by mask_sq
__global__ void k_maskchol(float* __restrict__ cov, const float* __restrict__ mask)
{
    long i = (long)blockIdx.x * blockDim.x + threadIdx.x;
    if (i >= (long)NB * NV * NL * NL) return;
    int mat = (int)(i / (NL * NL));
    int rem = (int)(i % (NL * NL));
    int l = rem / NL, k = rem - l * NL;
    int b = mat >> 6;
    float v = (k <= l) ? cov[i] * mask[b * NL + l] * mask[b * NL + k] : 0.f;
    cov[i] = v;
}

// samples[b,s,l, h*NV+d] = mean[h,b,l,d] + sum_j cholc[b,d,l,j] * eps[h,b,s,d,j]
__global__ void k_samp(const float* __restrict__ cholc, const float* __restrict__ eps_h,
                       const float* __restrict__ mean_h, float* __restrict__ samples, int h)
{
    int idx = blockIdx.x;          // (b,s,d)
    int d = idx & 63;
    int s = (idx >> 6) & 3;
    int b = idx >> 8;
    int l = threadIdx.x;
    const float* Cc = cholc + (long)(b * NV + d) * NL * NL + (long)l * NL;
    const float* E  = eps_h + (long)((b * NS + s) * NV + d) * NL;
    float acc = 0.f;
    for (int j = 0; j < NL; j++) acc += Cc[j] * E[j];
    float mv = mean_h[((long)b * NL + l) * NV + d];
    samples[(((long)(b * NS + s)) * NL + l) * NHD + h * NV + d] = mv + acc;
}

// =====================================================================
// KL reduction per head (one block), atomicAdd into out_kl
// =====================================================================
__global__ void k_zero(float* p) { p[0] = 0.f; }

__global__ void k_kl(const float* __restrict__ sloc, const float* __restrict__ lssd,
                     const float* __restrict__ vb, const float* __restrict__ Kbb,
                     const float* __restrict__ vg, const float* __restrict__ ubuf,
                     float* __restrict__ out_kl)
{
    int h = blockIdx.x, t = threadIdx.x;
    const float* S  = sloc + (long)h * NV * NM * NM;
    const float* LS = lssd + (long)h * NV * NM;
    const float* V  = vb   + (long)h * NM * NV;
    const float* Kb = Kbb  + (long)h * NM * NM;
    const float* VG = vg   + (long)h * NB * NL * NV;
    const float* U  = ubuf + (long)h * NB * NL * NV;
    float acc = 0.f;
    for (int i = t; i < NV * NM * NM; i += 256) { float v = S[i]; acc += 0.5f * v * v; }
    for (int i = t; i < NV * NM; i += 256) acc -= LS[i];
    for (int i = t; i < NM * NM; i += 256) {
        int m = i >> 6, n = i & 63;
        float dp = 0.f;
        for (int d = 0; d < NV; d++) dp += V[m * NV + d] * V[n * NV + d];
        acc += 0.5f * Kb[i] * dp;
    }
    for (int i = t; i < NB * NL * NV; i += 256) acc += (0.5f / NB) * VG[i] * U[i];
    __shared__ float red[256];
    red[t] = acc; __syncthreads();
    for (int s = 128; s > 0; s >>= 1) { if (t < s) red[t] += red[t + s]; __syncthreads(); }
    if (t == 0) atomicAdd(out_kl, red[0] - 0.5f * NM * NV);
}

__global__ void k_bias(float* __restrict__ out, const float* __restrict__ bias)
{
    long i = (long)blockIdx.x * blockDim.x + threadIdx.x;
    if (i >= (long)NB * NS * NL * NHD) return;
    out[i] += bias[i % NHD];
}

// =====================================================================
// Host side
// =====================================================================
static inline void gemm(hipStream_t st, int tA, int tB,
                        const float* A, int lda, long sA, int adiv, int amod,
                        const float* B, int ldb, long sB, int bdiv, int bmod,
                        float* C, int ldc, long sC,
                        const float* Cadd, long sCadd, int cadiv, int camod, float beta,
                        int M, int N, int K, float alpha, int batch)
{
    int tiles = (M / 16) * (N / 16);
    dim3 grid((tiles + 7) / 8, batch);
    if (tA == 0 && tB == 0)
        wmma_gemm<0,0><<<grid, 256, 0, st>>>(A,lda,sA,adiv,amod, B,ldb,sB,bdiv,bmod,
                                             C,ldc,sC, Cadd,sCadd,cadiv,camod,beta, M,N,K,alpha);
    else if (tA == 0 && tB == 1)
        wmma_gemm<0,1><<<grid, 256, 0, st>>>(A,lda,sA,adiv,amod, B,ldb,sB,bdiv,bmod,
                                             C,ldc,sC, Cadd,sCadd,cadiv,camod,beta, M,N,K,alpha);
    else if (tA == 1 && tB == 0)
        wmma_gemm<1,0><<<grid, 256, 0, st>>>(A,lda,sA,adiv,amod, B,ldb,sB,bdiv,bmod,
                                             C,ldc,sC, Cadd,sCadd,cadiv,camod,beta, M,N,K,alpha);
    else
        wmma_gemm<1,1><<<grid, 256, 0, st>>>(A,lda,sA,adiv,amod, B,ldb,sB,bdiv,bmod,
                                             C,ldc,sC, Cadd,sCadd,cadiv,camod,beta, M,N,K,alpha);
}

extern "C" void kernel_launch(void* const* d_in, const int* in_sizes, int n_in,
                              void* d_out, int out_size, void* d_ws, size_t ws_size,
                              hipStream_t stream)
{
    (void)in_sizes; (void)n_in; (void)out_size; (void)ws_size;
    const float* x      = (const float*)d_in[0];
    const float* cur_k  = (const float*)d_in[1];
    const float* mask   = (const float*)d_in[2];
    const float* eps    = (const float*)d_in[3];
    const float* Wqk    = (const float*)d_in[4];
    const float* Wv     = (const float*)d_in[5];
    const float* vparam = (const float*)d_in[6];
    const float* ltri   = (const float*)d_in[7];
    const float* lssd   = (const float*)d_in[8];
    const float* logls  = (const float*)d_in[9];
    const float* logsf  = (const float*)d_in[10];
    const float* WOw    = (const float*)d_in[11];
    const float* WOb    = (const float*)d_in[12];
    float* out = (float*)d_out;

    float* w = (float*)d_ws;
    float* q_all  = w;              // 384x512
    float* vg_raw = q_all  + 196608;
    float* kb     = vg_raw + 196608; // (h,m,v)
    float* qs     = kb     + 32768;  // (h,b,l,v)
    float* ks     = qs     + 196608;
    float* vg     = ks     + 32768;
    float* qn     = vg     + 196608;
    float* kn     = qn     + 3072;
    float* Kqq    = kn     + 512;    // (hb,L,L)
    float* Kbg    = Kqq    + 589824; // (hb,M,L)
    float* Kbb    = Kbg    + 196608;
    float* cholM  = Kbb    + 32768;
    float* v1     = cholM  + 32768;  // (hb,M,L)
    float* v2     = v1     + 196608;
    float* Abuf   = v2     + 196608; // (hb,L,L)
    float* meanb  = Abuf   + 589824; // (h,b,L,V)
    float* covar0 = meanb  + 196608; // (hb,L,L)
    float* Wbuf   = covar0 + 589824;
    float* ubuf   = Wbuf   + 589824; // (h,b,L,V)
    float* sloc   = ubuf   + 196608; // (h,d,M,M)
    float* samples= sloc   + 2097152;// (b,s,l,512)
    float* v3     = samples+ 786432; // per-head (b,d,L,M)
    float* covar  = v3     + 1572864;// per-head (b,d,L,L)

    // ---- projections (WMMA) ----
    gemm(stream, 0,1, x, NHD, 0, 1, BIGMOD,  Wqk, NHD, 0, 1, BIGMOD,
         q_all, NHD, 0, nullptr, 0, 1, BIGMOD, 0.f, NB*NL, NHD, NHD, 1.f, 1);
    gemm(stream, 0,1, x, NHD, 0, 1, BIGMOD,  Wv, NHD, 0, 1, BIGMOD,
         vg_raw, NHD, 0, nullptr, 0, 1, BIGMOD, 0.f, NB*NL, NHD, NHD, 1.f, 1);
    gemm(stream, 0,1, cur_k, NHD, (long)NM*NHD, 1, BIGMOD,  Wqk, NHD, (long)NV*NHD, 1, BIGMOD,
         kb, NV, (long)NM*NV, nullptr, 0, 1, BIGMOD, 0.f, NM, NV, NHD, 1.f, NH);

    // ---- ARD features / norms ----
    k_feat_q<<<(NH*NB*NL + 255)/256, 256, 0, stream>>>(q_all, vg_raw, mask, logls, qs, vg, qn);
    k_feat_k<<<(NH*NM + 255)/256, 256, 0, stream>>>(kb, logls, ks, kn);

    // ---- Gram matrices (WMMA) ----
    gemm(stream, 0,1, qs, NV, (long)NL*NV, 1, BIGMOD,  qs, NV, (long)NL*NV, 1, BIGMOD,
         Kqq, NL, (long)NL*NL, nullptr, 0, 1, BIGMOD, 0.f, NL, NL, NV, 1.f, NH*NB);
    gemm(stream, 0,1, ks, NV, (long)NM*NV, NB, BIGMOD,  qs, NV, (long)NL*NV, 1, BIGMOD,
         Kbg, NL, (long)NM*NL, nullptr, 0, 1, BIGMOD, 0.f, NM, NL, NV, 1.f, NH*NB);
    gemm(stream, 0,1, ks, NV, (long)NM*NV, 1, BIGMOD,  ks, NV, (long)NM*NV, 1, BIGMOD,
         Kbb, NM, (long)NM*NM, nullptr, 0, 1, BIGMOD, 0.f, NM, NM, NV, 1.f, NH);

    // ---- kernels from Grams ----
    k_make_qq<<<(NH*NB*NL*NL + 255)/256, 256, 0, stream>>>(Kqq, qn, mask, logsf);
    k_make_bg<<<(NH*NB*NM*NL + 255)/256, 256, 0, stream>>>(Kbg, kn, qn, mask, logsf);
    k_make_bb<<<(NH*NM*NM + 255)/256, 256, 0, stream>>>(Kbb, kn, logsf);

    // ---- 64x64 Cholesky + triangular solves ----
    k_chol64<<<NH, NM, 0, stream>>>(Kbb, cholM);
    { dim3 g(NH*NB, NL/NM); k_trisolve<<<g, NM, 0, stream>>>(cholM, Kbg, v1); }
    k_maskv2<<<(NH*NB*NM*NL + 255)/256, 256, 0, stream>>>(v1, mask, v2);

    // ---- Abuf = Kqq - Kqb@Kbg ; mean ; covar0 ; Wbuf ; ubuf (all WMMA) ----
    gemm(stream, 1,0, Kbg, NL, (long)NM*NL, 1, BIGMOD,  Kbg, NL, (long)NM*NL, 1, BIGMOD,
         Abuf, NL, (long)NL*NL, Kqq, (long)NL*NL, 1, BIGMOD, 1.f, NL, NL, NM, -1.f, NH*NB);
    gemm(stream, 0,0, Abuf, NL, (long)NL*NL, 1, BIGMOD,  vg, NV, (long)NL*NV, 1, BIGMOD,
         meanb, NV, (long)NL*NV, nullptr, 0, 1, BIGMOD, 0.f, NL, NV, NL, 1.f, NH*NB);
    gemm(stream, 1,0, Kbg, NL, (long)NM*NL, 1, BIGMOD,  vparam, NV, (long)NM*NV, NB, BIGMOD,
         meanb, NV, (long)NL*NV, meanb, (long)NL*NV, 1, BIGMOD, 1.f, NL, NV, NM, 1.f, NH*NB);
    gemm(stream, 1,0, v1, NL, (long)NM*NL, 1, BIGMOD,  v1, NL, (long)NM*NL, 1, BIGMOD,
         covar0, NL, (long)NL*NL, Kqq, (long)NL*NL, 1, BIGMOD, 1.f, NL, NL, NM, -1.f, NH*NB);
    gemm(stream, 1,0, v2, NL, (long)NM*NL, 1, BIGMOD,  v2, NL, (long)NM*NL, 1, BIGMOD,
         Wbuf, NL, (long)NL*NL, Kqq, (long)NL*NL, 1, BIGMOD, 1.f, NL, NL, NM, -1.f, NH*NB);
    gemm(stream, 0,0, Wbuf, NL, (long)NL*NL, 1, BIGMOD,  vg, NV, (long)NL*NV, 1, BIGMOD,
         ubuf, NV, (long)NL*NV, nullptr, 0, 1, BIGMOD, 0.f, NL, NV, NL, 1.f, NH*NB);

    // ---- s_local + KL ----
    k_sloc<<<(NH*NV*NM*NM + 255)/256, 256, 0, stream>>>(ltri, lssd, sloc);
    k_zero<<<1, 1, 0, stream>>>(out + KL_OFS);
    k_kl<<<NH, 256, 0, stream>>>(sloc, lssd, vparam, Kbb, vg, ubuf, out + KL_OFS);

    // ---- per-head: v3, covar, Cholesky(192), sampling ----
    for (int h = 0; h < NH; h++) {
        // v3[b,d] = v1[b]^T @ sloc[d]   (batch = NB*NV = 128)
        gemm(stream, 1,0, v1 + (long)h*NB*NM*NL, NL, (long)NM*NL, NV, BIGMOD,
             sloc + (long)h*NV*NM*NM, NM, (long)NM*NM, 1, NV,
             v3, NM, (long)NL*NM, nullptr, 0, 1, BIGMOD, 0.f, NL, NM, NM, 1.f, NB*NV);
        // covar[b,d] = covar0[b] + v3 @ v3^T
        gemm(stream, 0,1, v3, NM, (long)NL*NM, 1, BIGMOD,
             v3, NM, (long)NL*NM, 1, BIGMOD,
             covar, NL, (long)NL*NL, covar0 + (long)h*NB*NL*NL, (long)NL*NL, NV, BIGMOD, 1.f,
             NL, NL, NM, 1.f, NB*NV);
        k_prepdiag<<<NB*NV, NL, 0, stream>>>(covar, mask);
        k_chol192<<<NB*NV, NL, 0, stream>>>(covar);
        k_maskchol<<<(int)(((long)NB*NV*NL*NL + 255)/256), 256, 0, stream>>>(covar, mask);
        k_samp<<<NB*NS*NV, NL, 0, stream>>>(covar, eps + (long)h*NB*NS*NV*NL,
                                            meanb + (long)h*NB*NL*NV, samples, h);
    }

    // ---- output projection (WMMA) + bias ----
    gemm(stream, 0,1, samples, NHD, 0, 1, BIGMOD,  WOw, NHD, 0, 1, BIGMOD,
         out, NHD, 0, nullptr, 0, 1, BIGMOD, 0.f, NB*NS*NL, NHD, NHD, 1.f, 1);
    k_bias<<<(NB*NS*NL*NHD + 255)/256, 256, 0, stream>>>(out, WOb);
}